// SSRP_T_68032281968787
// MI455X (gfx1250) — compile-verified
//
#include <hip/hip_runtime.h>
#include <math.h>

// Problem constants (from reference): x (16,128,128,512) f32, W=4, K=12
#define B_  16
#define C_  128
#define F_  128
#define T_  512
#define ROWS (B_*C_*F_)         // 262144 rows of length T
#define ROWS_PER_WAVE 8
#define WAVES_PER_BLOCK 8       // blockDim = 256 = 8 wave32

typedef float v2f __attribute__((ext_vector_type(2)));
typedef float v8f __attribute__((ext_vector_type(8)));

// ---------------------------------------------------------------------------
// Async global -> LDS staging of one 2KiB row (512 f32). 32 lanes x 16B = 512B
// per instruction; 4 instructions cover the row. Tracked by ASYNCcnt.
// ---------------------------------------------------------------------------
__device__ __forceinline__ void async_row_to_lds(const float* gsrc, float* ldst, int lane) {
    const float* gp = gsrc + lane * 4;                       // 16B per lane
    unsigned lp = (unsigned)(unsigned long long)(ldst + lane * 4); // low 32b of flat LDS ptr = LDS offset
#pragma unroll
    for (int i = 0; i < 4; ++i) {
        unsigned lofs = lp + (unsigned)(i * 512);            // bytes
        const float* gofs = gp + i * 128;                    // 128 floats = 512B
        asm volatile("global_load_async_to_lds_b128 %0, %1, off"
                     :: "v"(lofs), "v"(gofs)
                     : "memory");
    }
}

// DPP lane exchange (VALU path; avoids LDS-permute traffic of __shfl).
template <int CTRL>
__device__ __forceinline__ float dpp_xchg(float v) {
    int s = __float_as_int(v);
    int r = __builtin_amdgcn_update_dpp(s, s, CTRL, 0xF, 0xF, false);
    return __int_as_float(r);
}
#define DPP_XOR1        0xB1    // quad_perm [1,0,3,2]  : pairs (l, l^1)
#define DPP_MIRROR4     0x1B    // quad_perm [3,2,1,0]  : mirror in groups of 4
#define DPP_HALF_MIRROR 0x141   // ROW_HALF_MIRROR      : mirror in groups of 8
#define DPP_MIRROR16    0x140   // ROW_MIRROR           : mirror in groups of 16

// One butterfly-merge level: both partner lanes end with the identical sorted
// (descending) top-16 of the union of their lists. `exch` must be an
// involution pairing the two halves of the current lane group.
template <typename X>
__device__ __forceinline__ void merge_level(float (&w)[16], X exch) {
    float o[16];
#pragma unroll
    for (int i = 0; i < 16; ++i) o[i] = exch(w[i]);
    float m[16];
#pragma unroll
    for (int i = 0; i < 16; ++i) m[i] = fmaxf(w[i], o[15 - i]);  // bitonic split -> top-16
    // bitonic clean, descending
#pragma unroll
    for (int j = 8; j > 0; j >>= 1) {
#pragma unroll
        for (int i = 0; i < 16; ++i) {
            int l = i ^ j;
            if (l > i) { float a = m[i], b = m[l]; m[i] = fmaxf(a, b); m[l] = fminf(a, b); }
        }
    }
#pragma unroll
    for (int i = 0; i < 16; ++i) w[i] = m[i];
}

// ---------------------------------------------------------------------------
// Per-row: window sums, per-lane bitonic sort-16, cross-lane butterfly merge.
// Returns (sum of top-12 window sums) / 48 == mean of top-12 window MEANS.
// (Only lane 0's return value is consumed.)
// ---------------------------------------------------------------------------
__device__ __forceinline__ float topk_row(const float* cur, int lane) {
    const float4* cp = (const float4*)(cur + lane * 16);
    float4 q0 = cp[0], q1 = cp[1], q2 = cp[2], q3 = cp[3];

    float xv[19];
    xv[0]=q0.x;  xv[1]=q0.y;  xv[2]=q0.z;  xv[3]=q0.w;
    xv[4]=q1.x;  xv[5]=q1.y;  xv[6]=q1.z;  xv[7]=q1.w;
    xv[8]=q2.x;  xv[9]=q2.y;  xv[10]=q2.z; xv[11]=q2.w;
    xv[12]=q3.x; xv[13]=q3.y; xv[14]=q3.z; xv[15]=q3.w;
    xv[16] = __shfl_down(q0.x, 1);   // crosses the row-16 boundary -> keep shfl
    xv[17] = __shfl_down(q0.y, 1);
    xv[18] = __shfl_down(q0.z, 1);

    // Rolling window sums: w[j] = x[16l+j] + ... + x[16l+j+3]
    float w[16];
    w[0] = xv[0] + xv[1] + xv[2] + xv[3];
#pragma unroll
    for (int j = 1; j < 16; ++j) w[j] = w[j-1] - xv[j-1] + xv[j+3];
    if (lane == 31) {                // windows 509..511 don't exist
        const float ninf = -__builtin_huge_valf();
        w[13] = ninf; w[14] = ninf; w[15] = ninf;
    }

    // Bitonic sort 16 values, descending (fully unrolled -> registers only).
#pragma unroll
    for (int k = 2; k <= 16; k <<= 1) {
#pragma unroll
        for (int j = k >> 1; j > 0; j >>= 1) {
#pragma unroll
            for (int i = 0; i < 16; ++i) {
                int l = i ^ j;
                if (l > i) {
                    bool desc = ((i & k) == 0);
                    float a = w[i], b = w[l];
                    float hi = fmaxf(a, b), lo = fminf(a, b);
                    w[i] = desc ? hi : lo;
                    w[l] = desc ? lo : hi;
                }
            }
        }
    }

    // Butterfly merge: DPP exchanges for the 4 intra-row-16 levels.
    merge_level(w, [](float v) { return dpp_xchg<DPP_XOR1>(v); });        // groups of 2
    merge_level(w, [](float v) { return dpp_xchg<DPP_MIRROR4>(v); });     // groups of 4
    merge_level(w, [](float v) { return dpp_xchg<DPP_HALF_MIRROR>(v); }); // groups of 8
    merge_level(w, [](float v) { return dpp_xchg<DPP_MIRROR16>(v); });    // groups of 16

    // Final level (crosses DPP16 rows): exchange via shfl, then skip the sort:
    // sum(top-12) = sum(bitonic top-16) - sum(4 smallest of it).
    float o[16];
#pragma unroll
    for (int i = 0; i < 16; ++i) o[i] = __shfl_xor(w[i], 16);
    float m[16];
#pragma unroll
    for (int i = 0; i < 16; ++i) m[i] = fmaxf(w[i], o[15 - i]);  // global top-16 (bitonic)

    float tot = 0.f;
#pragma unroll
    for (int i = 0; i < 16; ++i) tot += m[i];
    float a8[8];
#pragma unroll
    for (int i = 0; i < 8; ++i) a8[i] = fminf(m[i], m[i + 8]);   // 8 smallest (bitonic)
    float b4[4];
#pragma unroll
    for (int i = 0; i < 4; ++i) b4[i] = fminf(a8[i], a8[i + 4]); // 4 smallest
    float s4 = (b4[0] + b4[1]) + (b4[2] + b4[3]);

    return (tot - s4) * (1.0f / 48.0f);   // /W (window mean) and /K (top-k mean)
}

// ---------------------------------------------------------------------------
// Phase 1: one wave per row, 8 rows per wave, async double-buffered via LDS.
// ---------------------------------------------------------------------------
__global__ __launch_bounds__(256) void ssrp_phase1(const float* __restrict__ x,
                                                   float* __restrict__ z) {
    __shared__ float lbuf[WAVES_PER_BLOCK][2][T_];   // 32 KiB per workgroup
    const int lane = threadIdx.x & 31;
    const int wv   = threadIdx.x >> 5;
    const long wave_id = (long)blockIdx.x * WAVES_PER_BLOCK + wv;
    const long row0 = wave_id * ROWS_PER_WAVE;

    async_row_to_lds(x + row0 * T_, &lbuf[wv][0][0], lane);

#pragma unroll 1
    for (int k = 0; k < ROWS_PER_WAVE; ++k) {
        if (k + 1 < ROWS_PER_WAVE) {
            // make sure prior ds_reads of the target buffer have drained
            asm volatile("s_wait_dscnt 0x0" ::: "memory");
            async_row_to_lds(x + (row0 + k + 1) * T_, &lbuf[wv][(k + 1) & 1][0], lane);
            // 4 loads of current row complete (in-order) when <=4 outstanding
            asm volatile("s_wait_asynccnt 0x4" ::: "memory");
        } else {
            asm volatile("s_wait_asynccnt 0x0" ::: "memory");
        }
        float zz = topk_row(&lbuf[wv][k & 1][0], lane);
        if (lane == 0) z[row0 + k] = zz;
    }
}

// ---------------------------------------------------------------------------
// Phase 2: out[p] = mean_f z[p,f], p in [0,2048), f in [0,128).
// Row-sum on the matrix unit: V_WMMA_F32_16X16X4_F32 with B = all-ones makes
// every column of D the row sum over the 4-wide K chunk; accumulate 32 chunks.
// A 16x4 f32 layout: lanes 0-15 hold (M=lane, K=0/1), lanes 16-31 (M, K=2/3).
// C/D layout: lane N<16 VGPR v = D[M=v][N]; lane N+16 VGPR v = D[M=v+8][N].
// ---------------------------------------------------------------------------
__global__ __launch_bounds__(32) void ssrp_phase2(const float* __restrict__ z,
                                                  float* __restrict__ out) {
    const int lane = threadIdx.x & 31;
    const int p0 = blockIdx.x * 16;
    const int M  = lane & 15;
    const int kb = (lane >> 4) << 1;                 // 0 for lanes 0-15, 2 for 16-31
    const float* rp = z + (size_t)(p0 + M) * F_ + kb;

    v8f c = {0.f, 0.f, 0.f, 0.f, 0.f, 0.f, 0.f, 0.f};
    v2f bones = {1.0f, 1.0f};                        // B = ones(4,16)

#pragma unroll 4
    for (int f = 0; f < F_; f += 4) {
        v2f a;
        a.x = rp[f];
        a.y = rp[f + 1];
        c = __builtin_amdgcn_wmma_f32_16x16x4_f32(
                /*neg_a=*/false, a, /*neg_b=*/false, bones,
                /*c_mod=*/(short)0, c, /*reuse_a=*/false, /*reuse_b=*/false);
    }

    // Column N=0 of D carries the sums: lane 0 has M=0..7, lane 16 has M=8..15.
    if (lane == 0 || lane == 16) {
        const int mbase = (lane == 0) ? 0 : 8;
#pragma unroll
        for (int v = 0; v < 8; ++v)
            out[p0 + mbase + v] = c[v] * (1.0f / (float)F_);
    }
}

// ---------------------------------------------------------------------------
extern "C" void kernel_launch(void* const* d_in, const int* in_sizes, int n_in,
                              void* d_out, int out_size, void* d_ws, size_t ws_size,
                              hipStream_t stream) {
    const float* x = (const float*)d_in[0];    // (16,128,128,512) f32
    float* out = (float*)d_out;                // (16,128) f32 = 2048
    float* zws = (float*)d_ws;                 // needs ROWS*4 = 1 MiB scratch

    // ROWS / (ROWS_PER_WAVE * WAVES_PER_BLOCK) = 262144 / 64 = 4096 blocks
    dim3 g1(ROWS / (ROWS_PER_WAVE * WAVES_PER_BLOCK)), b1(256);
    ssrp_phase1<<<g1, b1, 0, stream>>>(x, zws);

    dim3 g2((B_ * C_) / 16), b2(32);           // 128 tiles of 16 output rows
    ssrp_phase2<<<g2, b2, 0, stream>>>(zws, out);
}